// SAGENet_4964982194740
// MI455X (gfx1250) — compile-verified
//
#include <hip/hip_runtime.h>
#include <hip/hip_bf16.h>

typedef __attribute__((ext_vector_type(2))) float v2f;
typedef __attribute__((ext_vector_type(8))) float v8f;

#define IN_CH 128
#define HID   16
#define OUT_CH 64
#define N1_NODES 100000

// ---------------------------------------------------------------------------
// GEMM1: y[N0,16] = x[N0,128] @ W1[128,16]   (one wave per 16-row tile)
// A layout (f32 16x4): lanes 0-15 -> M=lane, VGPR0=K0,VGPR1=K1; lanes 16-31 -> K2,K3
// B layout (f32 4x16): mirrored K-halves, N striped across lanes 0-15
// ---------------------------------------------------------------------------
__global__ void k_gemm_xw1(const float* __restrict__ x, const float* __restrict__ W1,
                           float* __restrict__ y, int nrows) {
  int wave = blockIdx.x * (blockDim.x >> 5) + (threadIdx.x >> 5);
  int ntiles = nrows >> 4;
  if (wave >= ntiles) return;                 // wave-uniform: EXEC stays all-ones
  int lane = threadIdx.x & 31;
  int half = lane >> 4;                       // 0: K pair {0,1}, 1: K pair {2,3}
  int mn   = lane & 15;                       // M index for A, N index for B
  int r0   = wave << 4;
  const float2* arow = (const float2*)(x + (size_t)(r0 + mn) * IN_CH);
  v8f c = {};
  for (int k0 = 0; k0 < IN_CH; k0 += 4) {
    int kb = k0 + 2 * half;
    float2 av = arow[(k0 >> 1) + half];
    v2f a = {av.x, av.y};
    v2f b = {W1[kb * HID + mn], W1[(kb + 1) * HID + mn]};
    c = __builtin_amdgcn_wmma_f32_16x16x4_f32(false, a, false, b, (short)0, c,
                                              false, false);
  }
  // D layout: VGPR j, lanes 0-15 -> M=j; lanes 16-31 -> M=j+8; N=lane&15
  float* yo = y + ((size_t)(r0 + 8 * half)) * HID + mn;
#pragma unroll
  for (int j = 0; j < 8; ++j) yo[(size_t)j * HID] = c[j];
}

// ---------------------------------------------------------------------------
// Edge scatter: acc[dst] += y[src] (16 f32 hardware atomics), cnt[dst] += 1
// ---------------------------------------------------------------------------
__global__ void k_scatter(const float* __restrict__ feat, const int* __restrict__ src,
                          const int* __restrict__ dst, float* __restrict__ acc,
                          float* __restrict__ cnt, int nedges) {
  int e = blockIdx.x * blockDim.x + threadIdx.x;
  if (e >= nedges) return;
  int s = src[e], t = dst[e];
  const float4* fv = (const float4*)(feat + (size_t)s * HID);
  float4 a0 = fv[0], a1 = fv[1], a2 = fv[2], a3 = fv[3];
  float* d = acc + (size_t)t * HID;
  unsafeAtomicAdd(d + 0,  a0.x); unsafeAtomicAdd(d + 1,  a0.y);
  unsafeAtomicAdd(d + 2,  a0.z); unsafeAtomicAdd(d + 3,  a0.w);
  unsafeAtomicAdd(d + 4,  a1.x); unsafeAtomicAdd(d + 5,  a1.y);
  unsafeAtomicAdd(d + 6,  a1.z); unsafeAtomicAdd(d + 7,  a1.w);
  unsafeAtomicAdd(d + 8,  a2.x); unsafeAtomicAdd(d + 9,  a2.y);
  unsafeAtomicAdd(d + 10, a2.z); unsafeAtomicAdd(d + 11, a2.w);
  unsafeAtomicAdd(d + 12, a3.x); unsafeAtomicAdd(d + 13, a3.y);
  unsafeAtomicAdd(d + 14, a3.z); unsafeAtomicAdd(d + 15, a3.w);
  unsafeAtomicAdd(cnt + t, 1.0f);
}

// mean + bias + relu  (layer-1 finalize)
__global__ void k_finalize1(const float* __restrict__ acc, const float* __restrict__ cnt,
                            const float* __restrict__ b1, float* __restrict__ h1, int n) {
  int i = blockIdx.x * blockDim.x + threadIdx.x;
  if (i >= n) return;
  int t = i >> 4, f = i & 15;
  float c = fmaxf(cnt[t], 1.0f);
  float v = acc[i] / c + b1[f];
  h1[i] = fmaxf(v, 0.0f);
}

// mean only (layer-2 finalize)
__global__ void k_finalize2(const float* __restrict__ acc, const float* __restrict__ cnt,
                            float* __restrict__ agg, int n) {
  int i = blockIdx.x * blockDim.x + threadIdx.x;
  if (i >= n) return;
  int t = i >> 4;
  agg[i] = acc[i] / fmaxf(cnt[t], 1.0f);
}

// ---------------------------------------------------------------------------
// GEMM2: out[N2,64] = agg[N2,16] @ W2[16,64] + b2   (wave per 16x16 tile)
// ---------------------------------------------------------------------------
__global__ void k_gemm2(const float* __restrict__ agg, const float* __restrict__ W2,
                        const float* __restrict__ b2, float* __restrict__ out, int nrows) {
  int wave = blockIdx.x * (blockDim.x >> 5) + (threadIdx.x >> 5);
  int total = (nrows >> 4) * (OUT_CH / 16);
  if (wave >= total) return;
  int rt = wave >> 2, ct = wave & 3;
  int lane = threadIdx.x & 31, half = lane >> 4, mn = lane & 15;
  int r0 = rt << 4, c0 = ct << 4;
  v8f c = {};
  for (int k0 = 0; k0 < HID; k0 += 4) {
    int kb = k0 + 2 * half;
    const float* ar = agg + (size_t)(r0 + mn) * HID + kb;
    v2f a = {ar[0], ar[1]};
    v2f b = {W2[kb * OUT_CH + c0 + mn], W2[(kb + 1) * OUT_CH + c0 + mn]};
    c = __builtin_amdgcn_wmma_f32_16x16x4_f32(false, a, false, b, (short)0, c,
                                              false, false);
  }
  float bias = b2[c0 + mn];
  float* o = out + (size_t)(r0 + 8 * half) * OUT_CH + c0 + mn;
#pragma unroll
  for (int j = 0; j < 8; ++j) o[(size_t)j * OUT_CH] = c[j] + bias;
}

// ---------------------------------------------------------------------------
// In-place log_softmax over rows of 64, one wave32 per row, 2 cols per lane
// ---------------------------------------------------------------------------
__global__ void k_logsoftmax(float* __restrict__ out, int nrows) {
  int wave = blockIdx.x * (blockDim.x >> 5) + (threadIdx.x >> 5);
  if (wave >= nrows) return;
  int lane = threadIdx.x & 31;
  float* row = out + (size_t)wave * OUT_CH;
  float v0 = row[lane], v1 = row[lane + 32];
  float m = fmaxf(v0, v1);
#pragma unroll
  for (int off = 16; off >= 1; off >>= 1) m = fmaxf(m, __shfl_xor(m, off, 32));
  float s = __expf(v0 - m) + __expf(v1 - m);
#pragma unroll
  for (int off = 16; off >= 1; off >>= 1) s += __shfl_xor(s, off, 32);
  float lse = m + __logf(s);
  row[lane] = v0 - lse;
  row[lane + 32] = v1 - lse;
}

extern "C" void kernel_launch(void* const* d_in, const int* in_sizes, int n_in,
                              void* d_out, int out_size, void* d_ws, size_t ws_size,
                              hipStream_t stream) {
  const float* x   = (const float*)d_in[0];
  const float* W1  = (const float*)d_in[1];
  const float* b1  = (const float*)d_in[2];
  const float* W2  = (const float*)d_in[3];
  const float* b2  = (const float*)d_in[4];
  const int* src1  = (const int*)d_in[5];
  const int* dst1  = (const int*)d_in[6];
  const int* src2  = (const int*)d_in[7];
  const int* dst2  = (const int*)d_in[8];
  float* out = (float*)d_out;

  const int N0 = in_sizes[0] / IN_CH;   // 500000
  const int E1 = in_sizes[5];           // 2000000
  const int E2 = in_sizes[7];           // 400000
  const int N1 = N1_NODES;              // 100000 (reference constant)
  const int N2 = out_size / OUT_CH;     // 10000

  // workspace layout (floats): y | h1 | agg2 | [acc1 cnt1 acc2 cnt2] (zeroed)
  float* w    = (float*)d_ws;
  float* y    = w;                    size_t off = (size_t)N0 * HID;
  float* h1   = w + off;              off += (size_t)N1 * HID;
  float* agg2 = w + off;              off += (size_t)N2 * HID;
  float* acc1 = w + off;              off += (size_t)N1 * HID;
  float* cnt1 = w + off;              off += (size_t)N1;
  float* acc2 = w + off;              off += (size_t)N2 * HID;
  float* cnt2 = w + off;              off += (size_t)N2;

  size_t zero_bytes = ((size_t)N1 * HID + N1 + (size_t)N2 * HID + N2) * sizeof(float);
  hipMemsetAsync(acc1, 0, zero_bytes, stream);

  const int BLK = 256;                      // 8 wave32 per block
  const int WPB = BLK / 32;

  // 1) y = x @ W1 via WMMA
  int tiles1 = N0 / 16;
  k_gemm_xw1<<<(tiles1 + WPB - 1) / WPB, BLK, 0, stream>>>(x, W1, y, N0);
  // 2) scatter layer 1
  k_scatter<<<(E1 + BLK - 1) / BLK, BLK, 0, stream>>>(y, src1, dst1, acc1, cnt1, E1);
  // 3) h1 = relu(acc1/cnt1 + b1)
  k_finalize1<<<(N1 * HID + BLK - 1) / BLK, BLK, 0, stream>>>(acc1, cnt1, b1, h1, N1 * HID);
  // 4) scatter layer 2
  k_scatter<<<(E2 + BLK - 1) / BLK, BLK, 0, stream>>>(h1, src2, dst2, acc2, cnt2, E2);
  // 5) agg2 = acc2/cnt2
  k_finalize2<<<(N2 * HID + BLK - 1) / BLK, BLK, 0, stream>>>(acc2, cnt2, agg2, N2 * HID);
  // 6) out = agg2 @ W2 + b2 via WMMA
  int waves2 = (N2 / 16) * (OUT_CH / 16);
  k_gemm2<<<(waves2 + WPB - 1) / WPB, BLK, 0, stream>>>(agg2, W2, b2, out, N2);
  // 7) in-place log_softmax
  k_logsoftmax<<<(N2 + WPB - 1) / WPB, BLK, 0, stream>>>(out, N2);
}